// scaleEqNonlinMaxp_88459146428602
// MI455X (gfx1250) — compile-verified
//
#include <hip/hip_runtime.h>
#include <math.h>

// ---------------------------------------------------------------------------
// MI455X (gfx1250) implementation: every FFT/low-pass stage is a batched small
// DFT matmul executed on V_WMMA_F32_16X16X4_F32 (fp32 tensor path; 57 cascaded
// transforms through GELUs need fp32). One workgroup = one (image, resolution)
// pipeline, fully resident in a 224 KB LDS arena (WGP has 320 KB).
// Twiddles + Dirichlet low-pass kernels generated on the fly (TRANS pipe
// co-executes with XDL WMMA on CDNA5).
// NOTE: fp32 WMMA has no A/B NEG modifier (ISA 7.12: only C can be negated for
// F32), so the -Aim*Bim term negates the A fragment in VALU first.
// ---------------------------------------------------------------------------

typedef float v2f __attribute__((ext_vector_type(2)));
typedef float v8f __attribute__((ext_vector_type(8)));

#define PI2 6.28318530717958647692f
#define NO_WRAP 0x40000000

__device__ __forceinline__ v8f wmma_p(v2f a, v2f b, v8f c) {
  // D = A*B + C   (fp32 16x16x4 WMMA)
  return __builtin_amdgcn_wmma_f32_16x16x4_f32(false, a, false, b, (short)0, c,
                                               false, false);
}
__device__ __forceinline__ v8f wmma_n(v2f a, v2f b, v8f c) {
  // D = (-A)*B + C ; f32 WMMA has no neg_a modifier -> negate fragment in VALU
  v2f na = -a;
  return __builtin_amdgcn_wmma_f32_16x16x4_f32(false, na, false, b, (short)0, c,
                                               false, false);
}

// e^{i * sign2piP * f(fidx) * x} * scale ; packed-frequency mapping:
// f = fidx           (fidx <  rp)
// f = fidx - band    (fidx >= rp)   [negative-frequency rows packed at tail]
__device__ __forceinline__ float2 twiddle(int fidx, int x, float s2piP,
                                          int rp, int band, float scale) {
  int f = fidx - ((fidx >= rp) ? band : 0);
  float s, c;
  __sincosf(s2piP * (float)f * (float)x, &s, &c);
  return make_float2(c * scale, s * scale);
}

// Closed-form Dirichlet kernel of the crop-and-resynthesize low-pass:
// D[d] = (1/P) * sum_{k=-rn}^{rp-1} cos(2*pi*k*d/P)
__device__ __forceinline__ float dirichlet(int d, int P, int rp, int rn) {
  float th = (PI2 / (float)P) * (float)d;
  float sh = __sinf(0.5f * th);
  if (fabsf(sh) < 1e-5f) return (float)(rp + rn) / (float)P;
  return (__sinf(((float)rp - 0.5f) * th) + __sinf(((float)rn + 0.5f) * th)) /
         (2.0f * sh * (float)P);
}

__device__ __forceinline__ float gelu_exact(float x) {
  return 0.5f * x * (1.0f + erff(x * 0.70710678118654752f));
}

// ---------------------------------------------------------------------------
// Generic WMMA tile engine: C[M x N] = A[M x K] * B[K x N]
//   MODE 0: complex * complex -> complex        (4 WMMA / K-step)
//   MODE 1: complex * complex -> real part only (2 WMMA / K-step)
//   MODE 2: real * real -> real                 (1 WMMA / K-step)
// Loaders return float2 {re, im} and must zero-pad out-of-range indices.
// 8 waves per block; each wave owns whole 16x16 output tiles, K stepped by 4.
// EXEC is all-ones at every WMMA (lane-divergent code only inside loaders).
// ---------------------------------------------------------------------------
template <int MODE, class FA, class FB, class FS>
__device__ void gemm_engine(int M, int N, int K, FA la, FB lb, FS st) {
  const int lane = threadIdx.x & 31;
  const int wave = threadIdx.x >> 5;
  const int nWaves = blockDim.x >> 5;
  const int lo = lane & 15;
  const int hi = lane >> 4;
  const int kk = hi << 1;  // this lane's K sub-offset within a K-step of 4
  const int tM = (M + 15) >> 4, tN = (N + 15) >> 4;

  for (int t = wave; t < tM * tN; t += nWaves) {
    const int m0 = (t / tN) << 4;
    const int n0 = (t % tN) << 4;
    v8f cre = {0.f, 0.f, 0.f, 0.f, 0.f, 0.f, 0.f, 0.f};
    v8f cim = {0.f, 0.f, 0.f, 0.f, 0.f, 0.f, 0.f, 0.f};
    const int ma = m0 + lo;  // A row held by this lane (ISA 7.12.2 layout)
    const int nb = n0 + lo;  // B col held by this lane

    for (int k0 = 0; k0 < K; k0 += 4) {
      float2 a0 = la(ma, k0 + kk);
      float2 a1 = la(ma, k0 + kk + 1);
      float2 b0 = lb(k0 + kk, nb);
      float2 b1 = lb(k0 + kk + 1, nb);
      v2f are = {a0.x, a1.x};
      v2f bre = {b0.x, b1.x};
      cre = wmma_p(are, bre, cre);
      if (MODE < 2) {
        v2f aim = {a0.y, a1.y};
        v2f bim = {b0.y, b1.y};
        cre = wmma_n(aim, bim, cre);  // cre -= aim*bim
        if (MODE == 0) {
          cim = wmma_p(are, bim, cim);
          cim = wmma_p(aim, bre, cim);
        }
      }
    }
    // C/D layout: VGPR v -> M = m0 + v + 8*hi ; N = n0 + lane[3:0]
    for (int v = 0; v < 8; ++v) {
      int m = m0 + v + (hi << 3);
      if (m < M && nb < N) st(m, nb, cre[v], (MODE == 0) ? cim[v] : 0.0f);
    }
  }
}

// LDS arena: [0, 32768) floats = "T" region (128 KB), [32768, 57344) = "A"
// region (96 KB). Total 224 KB dynamic LDS per workgroup.
#define T_OFF 32768
#define SMEM_FLOATS (32768 + 24576)

// ---------------------------------------------------------------------------
// K0: full-image rfft2 (norm='forward').  Xspec[img][128][72] (65 cols used).
// ---------------------------------------------------------------------------
__global__ void k_fft128(const float* __restrict__ x, float2* __restrict__ Xspec) {
  extern __shared__ float smem[];
  float2* Tc = (float2*)smem;
  float* xa = smem + T_OFF;
  const int img = blockIdx.x;
  const float* src = x + (size_t)img * 128 * 128;
  for (int i = threadIdx.x; i < 128 * 128; i += blockDim.x) {
    if (i + (int)blockDim.x < 128 * 128)
      __builtin_prefetch(src + i + blockDim.x, 0, 3);  // global_prefetch_b8
    xa[i] = src[i];
  }
  __syncthreads();
  // U[k][n] = (1/128) sum_m e^{-2pi i k m/128} x[m][n]
  {
    auto la = [=](int k, int m) -> float2 {
      if (k >= 128 || m >= 128) return make_float2(0.f, 0.f);
      return twiddle(k, m, -PI2 / 128.f, NO_WRAP, 0, 1.f / 128.f);
    };
    auto lb = [=](int m, int n) -> float2 {
      if (m >= 128 || n >= 128) return make_float2(0.f, 0.f);
      return make_float2(xa[m * 128 + n], 0.f);
    };
    auto st = [=](int k, int n, float re, float im) {
      Tc[k * 128 + n] = make_float2(re, im);
    };
    gemm_engine<0>(128, 128, 128, la, lb, st);
  }
  __syncthreads();
  // Xspec[k][k2] = (1/128) sum_n U[k][n] e^{-2pi i n k2/128},  k2 < 65
  float2* dst = Xspec + (size_t)img * 128 * 72;
  {
    auto la = [=](int k, int n) -> float2 {
      if (k >= 128 || n >= 128) return make_float2(0.f, 0.f);
      return Tc[k * 128 + n];
    };
    auto lb = [=](int n, int k2) -> float2 {
      if (n >= 128 || k2 >= 65) return make_float2(0.f, 0.f);
      return twiddle(k2, n, -PI2 / 128.f, NO_WRAP, 0, 1.f / 128.f);
    };
    auto st = [=](int k, int k2, float re, float im) {
      dst[k * 72 + k2] = make_float2(re, im);
    };
    gemm_engine<0>(128, 65, 128, la, lb, st);
  }
}

// ---------------------------------------------------------------------------
// K1: full per-resolution pipeline in LDS.
//   crop(X,res) -> irfft2(res) -> GELU -> Dirichlet low-pass(ceil(res/2))
//   -> maxpool2 -> rfft2 -> shell scatter into x_out[img][64][36]
// ---------------------------------------------------------------------------
__global__ void k_res_stage(const float2* __restrict__ Xspec,
                            float2* __restrict__ xout, int res, int isBase) {
  extern __shared__ float smem[];
  float* Treg = smem;
  float* Areg = smem + T_OFF;
  const int img = blockIdx.x;

  const int rp1 = (res + 1) / 2;            // positive-freq rows incl DC
  const int c1 = res / 2 + 1;               // rfft half-spectrum cols
  const int band2 = (res + 1) / 2;          // ceil(res/POOL)
  const int rp2 = (band2 + 1) / 2, rn2 = band2 / 2;

  // Phase 0: stage cropped spectrum (packed rows: [0,rp1) pos, tail neg).
  float2* Xc = (float2*)Areg;
  const float2* Xs = Xspec + (size_t)img * 128 * 72;
  for (int i = threadIdx.x; i < res * c1; i += blockDim.x) {
    int j = i / c1, k2 = i - j * c1;
    int srcRow = (j < rp1) ? j : (128 - (res - j));
    Xc[i] = Xs[srcRow * 72 + k2];
  }
  __syncthreads();

  // Phase 1: column synthesis  T[j][n] = sum_{k2} w*Xc[j][k2] e^{+2pi i k2 n/res}
  float2* Tc = (float2*)Treg;
  {
    const int R = res, C1 = c1;
    auto la = [=](int j, int k2) -> float2 {
      if (j >= R || k2 >= C1) return make_float2(0.f, 0.f);
      return Xc[j * C1 + k2];
    };
    auto lb = [=](int k2, int n) -> float2 {
      if (k2 >= C1 || n >= R) return make_float2(0.f, 0.f);
      float w = (k2 == 0 || k2 == C1 - 1) ? 1.f : 2.f;  // Hermitian fold
      return twiddle(k2, n, PI2 / (float)R, NO_WRAP, 0, w);
    };
    auto st = [=](int j, int n, float re, float im) {
      Tc[j * R + n] = make_float2(re, im);
    };
    gemm_engine<0>(R, R, C1, la, lb, st);
  }
  __syncthreads();

  // Phase 2: row synthesis + exact GELU:  s1[m][n] = gelu(Re(sum_j e^{+..} T))
  float* s1 = Areg;
  {
    const int R = res, RP = rp1;
    auto la = [=](int m, int j) -> float2 {
      if (m >= R || j >= R) return make_float2(0.f, 0.f);
      return twiddle(j, m, PI2 / (float)R, RP, R, 1.f);
    };
    auto lb = [=](int j, int n) -> float2 {
      if (j >= R || n >= R) return make_float2(0.f, 0.f);
      return Tc[j * R + n];
    };
    auto st = [=](int m, int n, float re, float) { s1[m * R + n] = gelu_exact(re); };
    gemm_engine<1>(R, R, R, la, lb, st);
  }
  __syncthreads();

  // Phase 3/4: band-limit to band2 == D * s1 * D^T (closed-form Dirichlet).
  float* t2 = Treg;
  {
    const int R = res, P2 = rp2, N2 = rn2;
    auto la = [=](int m, int j) -> float2 {
      if (m >= R || j >= R) return make_float2(0.f, 0.f);
      return make_float2(dirichlet(m - j, R, P2, N2), 0.f);
    };
    auto lb = [=](int j, int n) -> float2 {
      if (j >= R || n >= R) return make_float2(0.f, 0.f);
      return make_float2(s1[j * R + n], 0.f);
    };
    auto st = [=](int m, int n, float re, float) { t2[m * R + n] = re; };
    gemm_engine<2>(R, R, R, la, lb, st);
  }
  __syncthreads();
  float* s2 = Areg;
  {
    const int R = res, P2 = rp2, N2 = rn2;
    auto la = [=](int m, int j) -> float2 {
      if (m >= R || j >= R) return make_float2(0.f, 0.f);
      return make_float2(t2[m * R + j], 0.f);
    };
    auto lb = [=](int j, int n) -> float2 {
      if (j >= R || n >= R) return make_float2(0.f, 0.f);
      return make_float2(dirichlet(n - j, R, P2, N2), 0.f);
    };
    auto st = [=](int m, int n, float re, float) { s2[m * R + n] = re; };
    gemm_engine<2>(R, R, R, la, lb, st);
  }
  __syncthreads();

  // Phase 5: 2x2 maxpool -> p[c][c]
  const int c = res / 2;
  float* p = Treg;
  for (int i = threadIdx.x; i < c * c; i += blockDim.x) {
    int r = i / c, q = i - r * c;
    float a = s2[(2 * r) * res + 2 * q], b = s2[(2 * r) * res + 2 * q + 1];
    float d = s2[(2 * r + 1) * res + 2 * q], e = s2[(2 * r + 1) * res + 2 * q + 1];
    p[i] = fmaxf(fmaxf(a, b), fmaxf(d, e));
  }
  __syncthreads();

  // Phase 6: forward row DFT  U[k][n] = (1/c) sum_m e^{-..} p[m][n]
  float2* U = (float2*)Areg;
  {
    const int C = c;
    auto la = [=](int k, int m) -> float2 {
      if (k >= C || m >= C) return make_float2(0.f, 0.f);
      return twiddle(k, m, -PI2 / (float)C, NO_WRAP, 0, 1.f / (float)C);
    };
    auto lb = [=](int m, int n) -> float2 {
      if (m >= C || n >= C) return make_float2(0.f, 0.f);
      return make_float2(p[m * C + n], 0.f);
    };
    auto st = [=](int k, int n, float re, float im) {
      U[k * C + n] = make_float2(re, im);
    };
    gemm_engine<0>(C, C, C, la, lb, st);
  }
  __syncthreads();

  // Phase 7: forward col DFT (half spectrum) PF[k][k2], k2 <= c/2
  const int c2 = c / 2 + 1;
  float2* PF = (float2*)Treg;
  {
    const int C = c, C2 = c2;
    auto la = [=](int k, int n) -> float2 {
      if (k >= C || n >= C) return make_float2(0.f, 0.f);
      return U[k * C + n];
    };
    auto lb = [=](int n, int k2) -> float2 {
      if (n >= C || k2 >= C2) return make_float2(0.f, 0.f);
      return twiddle(k2, n, -PI2 / (float)C, NO_WRAP, 0, 1.f / (float)C);
    };
    auto st = [=](int k, int k2, float re, float im) {
      PF[k * C2 + k2] = make_float2(re, im);
    };
    gemm_engine<0>(C, C2, C, la, lb, st);
  }
  __syncthreads();

  // Phase 8: shell scatter into x_out (this WG owns its image's slice).
  float2* xo = xout + (size_t)img * 64 * 36;
  const int h = c / 2;
  if (isBase) {
    for (int i = threadIdx.x; i < h * (h + 1); i += blockDim.x) {
      int r = i / (h + 1), q = i - r * (h + 1);
      xo[r * 36 + q] = PF[r * c2 + q];
      xo[(64 - h + r) * 36 + q] = PF[(c - h + r) * c2 + q];
    }
  } else if (c & 1) {
    for (int q = threadIdx.x; q <= h; q += blockDim.x) {
      xo[h * 36 + q] = PF[h * c2 + q];
      xo[(64 - h) * 36 + q] = PF[(c - h) * c2 + q];
      xo[q * 36 + h] = PF[q * c2 + h];
    }
    for (int r = threadIdx.x; r < h; r += blockDim.x)
      xo[(64 - h + r) * 36 + h] = PF[(c - h + r) * c2 + h];
  } else {
    for (int q = threadIdx.x; q <= h; q += blockDim.x)
      xo[(64 - h) * 36 + q] = PF[(c - h) * c2 + q];
    for (int r = threadIdx.x; r < h; r += blockDim.x) {
      xo[(64 - h + r) * 36 + h] = PF[(c - h + r) * c2 + h];
      xo[r * 36 + h] = PF[r * c2 + h];
    }
  }
}

// ---------------------------------------------------------------------------
// K2: y = irfft2(x_out, 64x64); out = fft2(y) -> interleaved (re, im).
// ---------------------------------------------------------------------------
__global__ void k_final(const float2* __restrict__ xout, float* __restrict__ out) {
  extern __shared__ float smem[];
  float* Treg = smem;
  float* Areg = smem + T_OFF;
  const int img = blockIdx.x;

  float2* Xo = (float2*)Areg;
  const float2* src = xout + (size_t)img * 64 * 36;
  for (int i = threadIdx.x; i < 64 * 33; i += blockDim.x) {
    int r = i / 33, q = i - r * 33;
    Xo[r * 33 + q] = src[r * 36 + q];
  }
  __syncthreads();

  float2* Tc = (float2*)Treg;
  {  // column synthesis with Hermitian fold weights
    auto la = [=](int j, int k2) -> float2 {
      if (j >= 64 || k2 >= 33) return make_float2(0.f, 0.f);
      return Xo[j * 33 + k2];
    };
    auto lb = [=](int k2, int n) -> float2 {
      if (k2 >= 33 || n >= 64) return make_float2(0.f, 0.f);
      float w = (k2 == 0 || k2 == 32) ? 1.f : 2.f;
      return twiddle(k2, n, PI2 / 64.f, NO_WRAP, 0, w);
    };
    auto st = [=](int j, int n, float re, float im) {
      Tc[j * 64 + n] = make_float2(re, im);
    };
    gemm_engine<0>(64, 64, 33, la, lb, st);
  }
  __syncthreads();

  float* y = Areg;
  {  // row synthesis, real part (row freqs periodic -> identity map)
    auto la = [=](int m, int j) -> float2 {
      if (m >= 64 || j >= 64) return make_float2(0.f, 0.f);
      return twiddle(j, m, PI2 / 64.f, NO_WRAP, 0, 1.f);
    };
    auto lb = [=](int j, int n) -> float2 {
      if (j >= 64 || n >= 64) return make_float2(0.f, 0.f);
      return Tc[j * 64 + n];
    };
    auto st = [=](int m, int n, float re, float) { y[m * 64 + n] = re; };
    gemm_engine<1>(64, 64, 64, la, lb, st);
  }
  __syncthreads();

  float2* U2 = (float2*)Treg;
  {  // forward row DFT, 1/64
    auto la = [=](int k, int m) -> float2 {
      if (k >= 64 || m >= 64) return make_float2(0.f, 0.f);
      return twiddle(k, m, -PI2 / 64.f, NO_WRAP, 0, 1.f / 64.f);
    };
    auto lb = [=](int m, int n) -> float2 {
      if (m >= 64 || n >= 64) return make_float2(0.f, 0.f);
      return make_float2(y[m * 64 + n], 0.f);
    };
    auto st = [=](int k, int n, float re, float im) {
      U2[k * 64 + n] = make_float2(re, im);
    };
    gemm_engine<0>(64, 64, 64, la, lb, st);
  }
  __syncthreads();

  float* dst = out + (size_t)img * 64 * 64 * 2;
  {  // forward col DFT (full complex), 1/64; write stacked (re, im)
    auto la = [=](int k, int n) -> float2 {
      if (k >= 64 || n >= 64) return make_float2(0.f, 0.f);
      return U2[k * 64 + n];
    };
    auto lb = [=](int n, int k2) -> float2 {
      if (n >= 64 || k2 >= 64) return make_float2(0.f, 0.f);
      return twiddle(k2, n, -PI2 / 64.f, NO_WRAP, 0, 1.f / 64.f);
    };
    auto st = [=](int k, int k2, float re, float im) {
      dst[(k * 64 + k2) * 2 + 0] = re;
      dst[(k * 64 + k2) * 2 + 1] = im;
    };
    gemm_engine<0>(64, 64, 64, la, lb, st);
  }
}

__global__ void k_zero(float* __restrict__ p, size_t n) {
  size_t i = (size_t)blockIdx.x * blockDim.x + threadIdx.x;
  size_t st = (size_t)gridDim.x * blockDim.x;
  for (; i < n; i += st) p[i] = 0.f;
}

// ---------------------------------------------------------------------------
extern "C" void kernel_launch(void* const* d_in, const int* in_sizes, int n_in,
                              void* d_out, int out_size, void* d_ws,
                              size_t ws_size, hipStream_t stream) {
  (void)in_sizes; (void)n_in; (void)out_size; (void)ws_size;
  const float* x = (const float*)d_in[0];
  float* out = (float*)d_out;
  float* ws = (float*)d_ws;

  const int NIMG = 32 * 64;  // 2048 independent (B,C) images
  float2* Xspec = (float2*)ws;                                   // [2048][128][72]
  float2* xo = (float2*)(ws + (size_t)NIMG * 128 * 72 * 2);      // [2048][64][36]

  const size_t SMEM = (size_t)SMEM_FLOATS * sizeof(float);  // 224 KB/WGP
  (void)hipFuncSetAttribute((const void*)k_fft128,
                            hipFuncAttributeMaxDynamicSharedMemorySize,
                            (int)SMEM);
  (void)hipFuncSetAttribute((const void*)k_res_stage,
                            hipFuncAttributeMaxDynamicSharedMemorySize,
                            (int)SMEM);
  (void)hipFuncSetAttribute((const void*)k_final,
                            hipFuncAttributeMaxDynamicSharedMemorySize,
                            (int)SMEM);

  // Zero the accumulated output spectrum (harness poisons ws once).
  k_zero<<<2048, 256, 0, stream>>>((float*)xo, (size_t)NIMG * 64 * 36 * 2);

  // Global 128x128 rfft2 of every image (fp32 WMMA DFT matmuls).
  k_fft128<<<NIMG, 256, SMEM, stream>>>(x, Xspec);

  // Base branch then ascending resolutions; stream order reproduces the
  // reference's later-shell-overwrites-earlier semantics.
  k_res_stage<<<NIMG, 256, SMEM, stream>>>(Xspec, xo, 16, 1);
  for (int res = 18; res <= 128; res += 2)
    k_res_stage<<<NIMG, 256, SMEM, stream>>>(Xspec, xo, res, 0);

  // irfft2(64) + fft2(64) -> stacked (real, imag) output.
  k_final<<<NIMG, 256, SMEM, stream>>>(xo, out);
}